// PhysicsInformedNN_49005576847706
// MI455X (gfx1250) — compile-verified
//
#include <hip/hip_runtime.h>
#include <hip/hip_bf16.h>
#include <math.h>

typedef float v2f __attribute__((ext_vector_type(2)));
typedef float v8f __attribute__((ext_vector_type(8)));

#define D_IN   100          // true hidden width
#define P      112          // padded hidden width (7 * 16)
#define JT     7            // N-tiles of 16
#define KT     28           // K-steps of 4
#define TILES  4            // point tiles of 16 per block
#define PTS    64           // points per block
#define NTHR   512          // 16 waves

#define PI_F   3.14159265358979323846f
#define CX     0.2f                 // d(scaled x)/dx = 2/10
#define CT     (4.0f / PI_F)        // d(scaled t)/dt = 2/(pi/2)

__global__ __launch_bounds__(NTHR, 1)
void pinn_residual_kernel(const float* __restrict__ x, const float* __restrict__ t,
                          const float* __restrict__ W1, const float* __restrict__ b1,
                          const float* __restrict__ W2, const float* __restrict__ b2,
                          const float* __restrict__ W3, const float* __restrict__ b3,
                          const float* __restrict__ W4, const float* __restrict__ b4,
                          const float* __restrict__ W5, const float* __restrict__ b5,
                          float* __restrict__ out, int N)
{
    // Weights stored PAIR-INTERLEAVED: w_lds[(p*P + n)*2 + o] = W[2p+o][n]
    // (p = K-row pair 0..55, n = output neuron, o = 0/1) so each WMMA B operand
    // is ONE aligned ds_load_b64 straight into an even VGPR pair.
    __shared__ float w_lds[P * P];              // 50176 B
    __shared__ float bias_lds[P];               //   448 B
    __shared__ float h_lds[TILES][4][16 * P];   // 114688 B
    __shared__ float zx_lds[TILES][16 * P];     // 28672 B
    __shared__ float sbuf[PTS][4][2];           //  2048 B

    const int tid    = threadIdx.x;
    const int lane   = tid & 31;
    const int wave   = tid >> 5;
    const int tileId = wave >> 2;   // 0..3 : point tile
    const int ch     = wave & 3;    // 0..3 : dual-number channel
    const int half   = lane >> 4;   // 0/1  : half-wave
    const int r      = lane & 15;

    const int base_pt = blockIdx.x * PTS;

    // ---------------- Layer 1 (2 -> 100), VALU, seeds the 4 dual channels -------------
    for (int i = tid; i < PTS * P; i += NTHR) {
        int pt = i / P, n = i % P;
        int tI = pt >> 4, pl = pt & 15;
        float hv = 0.f, hx = 0.f, ht = 0.f, hxx = 0.f;
        if (n < D_IN) {
            int g = base_pt + pt;
            float xv = (g < N) ? x[g] : 0.0f;
            float tv = (g < N) ? t[g] : 0.0f;
            float sx = CX * (xv + 5.0f) - 1.0f;   // 2*(x+5)/10 - 1
            float st = CT * tv - 1.0f;            // 2*t/(pi/2) - 1
            float w0 = W1[n], w1 = W1[D_IN + n];
            float z  = sx * w0 + st * w1 + b1[n];
            float zx = CX * w0;
            float zt = CT * w1;
            float h  = tanhf(z);
            float gg = 1.0f - h * h;
            hv = h; hx = gg * zx; ht = gg * zt; hxx = -2.0f * h * gg * zx * zx;
        }
        h_lds[tI][0][pl * P + n] = hv;
        h_lds[tI][1][pl * P + n] = hx;
        h_lds[tI][2][pl * P + n] = ht;
        h_lds[tI][3][pl * P + n] = hxx;
    }

    // ---------------- Layers 2..4 (100 -> 100) via fp32 WMMA ---------------------------
    for (int l = 0; l < 3; ++l) {
        const float* Wg = (l == 0) ? W2 : (l == 1) ? W3 : W4;
        const float* bg = (l == 0) ? b2 : (l == 1) ? b3 : b4;

        // stage padded weights, pair-interleaved along K
        for (int i = tid; i < P * P; i += NTHR) {
            int o   = i & 1;
            int idx = i >> 1;          // p*P + n
            int p   = idx / P;
            int n   = idx - p * P;
            int k   = 2 * p + o;
            w_lds[i] = (k < D_IN && n < D_IN) ? Wg[k * D_IN + n] : 0.0f;
        }
        for (int i = tid; i < P; i += NTHR)
            bias_lds[i] = (i < D_IN) ? bg[i] : 0.0f;
        __syncthreads();

        // GEMM: this wave's channel tile (16 pts x 112) @ W (112 x 112)
        const float* hA = &h_lds[tileId][ch][0];
        v8f acc[JT];
#pragma unroll
        for (int j = 0; j < JT; ++j)
            acc[j] = (v8f){0.f, 0.f, 0.f, 0.f, 0.f, 0.f, 0.f, 0.f};

        for (int k = 0; k < KT; ++k) {
            int k0 = k * 4 + half * 2;       // even
            int p  = k * 2 + half;           // K-row pair index
            v2f a = *(const v2f*)&hA[r * P + k0];            // one ds_load_b64
#pragma unroll
            for (int j = 0; j < JT; ++j) {
                v2f b = *(const v2f*)&w_lds[(p * P + j * 16 + r) * 2];  // one ds_load_b64
                acc[j] = __builtin_amdgcn_wmma_f32_16x16x4_f32(
                    false, a, false, b, (short)0, acc[j], false, false);
            }
        }
        __syncthreads();   // all waves done reading old h and w_lds

        // phase 1: ch0 -> h' = tanh(z + b); ch1 -> park raw z_x in scratch
        if (ch == 0) {
#pragma unroll
            for (int j = 0; j < JT; ++j)
#pragma unroll
                for (int i = 0; i < 8; ++i) {
                    int pt = i + 8 * half, n = j * 16 + r;
                    float z = acc[j][i] + bias_lds[n];
                    h_lds[tileId][0][pt * P + n] = tanhf(z);   // pad cols: z=0 -> 0
                }
        } else if (ch == 1) {
#pragma unroll
            for (int j = 0; j < JT; ++j)
#pragma unroll
                for (int i = 0; i < 8; ++i) {
                    int pt = i + 8 * half, n = j * 16 + r;
                    zx_lds[tileId][pt * P + n] = acc[j][i];
                }
        }
        __syncthreads();

        // phase 2: derivative channels use g = 1 - h'^2
        if (ch == 1) {
#pragma unroll
            for (int j = 0; j < JT; ++j)
#pragma unroll
                for (int i = 0; i < 8; ++i) {
                    int pt = i + 8 * half, n = j * 16 + r;
                    float h = h_lds[tileId][0][pt * P + n];
                    float g = 1.0f - h * h;
                    h_lds[tileId][1][pt * P + n] = g * acc[j][i];
                }
        } else if (ch == 2) {
#pragma unroll
            for (int j = 0; j < JT; ++j)
#pragma unroll
                for (int i = 0; i < 8; ++i) {
                    int pt = i + 8 * half, n = j * 16 + r;
                    float h = h_lds[tileId][0][pt * P + n];
                    float g = 1.0f - h * h;
                    h_lds[tileId][2][pt * P + n] = g * acc[j][i];
                }
        } else if (ch == 3) {
#pragma unroll
            for (int j = 0; j < JT; ++j)
#pragma unroll
                for (int i = 0; i < 8; ++i) {
                    int pt = i + 8 * half, n = j * 16 + r;
                    float h  = h_lds[tileId][0][pt * P + n];
                    float g  = 1.0f - h * h;
                    float zx = zx_lds[tileId][pt * P + n];
                    h_lds[tileId][3][pt * P + n] = g * acc[j][i] - 2.0f * h * g * zx * zx;
                }
        }
        __syncthreads();
    }

    // ---------------- Layer 5 (100 -> 2), VALU dot products ----------------------------
    if (tid < PTS * 4) {
        int pt = tid >> 2, c = tid & 3;
        int tI = pt >> 4, pl = pt & 15;
        const float* hh = &h_lds[tI][c][pl * P];
        float s0 = 0.f, s1 = 0.f;
        for (int k = 0; k < D_IN; ++k) {
            float hk = hh[k];
            s0 = fmaf(hk, W5[k * 2 + 0], s0);
            s1 = fmaf(hk, W5[k * 2 + 1], s1);
        }
        sbuf[pt][c][0] = s0;
        sbuf[pt][c][1] = s1;
    }
    __syncthreads();

    // ---------------- PDE residual combine ---------------------------------------------
    if (tid < PTS) {
        int pt = tid;
        int g = base_pt + pt;
        if (g < N) {
            float u    = sbuf[pt][0][0] + b5[0];
            float v    = sbuf[pt][0][1] + b5[1];
            float u_t  = sbuf[pt][2][0];
            float v_t  = sbuf[pt][2][1];
            float u_xx = sbuf[pt][3][0];
            float v_xx = sbuf[pt][3][1];
            float sq   = u * u + v * v;
            out[g * 2 + 0] = u_t + 0.5f * v_xx + sq * v;   // f_u
            out[g * 2 + 1] = v_t - 0.5f * u_xx - sq * u;   // f_v
        }
    }
}

extern "C" void kernel_launch(void* const* d_in, const int* in_sizes, int n_in,
                              void* d_out, int out_size, void* d_ws, size_t ws_size,
                              hipStream_t stream) {
    const float* x  = (const float*)d_in[0];
    const float* t  = (const float*)d_in[1];
    const float* W1 = (const float*)d_in[2];
    const float* b1 = (const float*)d_in[3];
    const float* W2 = (const float*)d_in[4];
    const float* b2 = (const float*)d_in[5];
    const float* W3 = (const float*)d_in[6];
    const float* b3 = (const float*)d_in[7];
    const float* W4 = (const float*)d_in[8];
    const float* b4 = (const float*)d_in[9];
    const float* W5 = (const float*)d_in[10];
    const float* b5 = (const float*)d_in[11];
    float* out = (float*)d_out;
    int N = in_sizes[0];
    int blocks = (N + PTS - 1) / PTS;
    pinn_residual_kernel<<<blocks, NTHR, 0, stream>>>(
        x, t, W1, b1, W2, b2, W3, b3, W4, b4, W5, b5, out, N);
}